// DivEncLayer_44908178047418
// MI455X (gfx1250) — compile-verified
//
#include <hip/hip_runtime.h>

// Problem constants (match reference)
#define B_TOT 32768
#define Q_N   128
#define V_N   8
#define H_N   32
#define LN_EPS 1e-5f
#define TILES 4            // 16-row batch tiles per block, double-buffered

typedef __attribute__((ext_vector_type(2))) float v2f;
typedef __attribute__((ext_vector_type(4))) float v4f;
typedef __attribute__((ext_vector_type(8))) float v8f;

// One DPP16 butterfly step: v += dpp_permute(v).
template <int CTRL>
__device__ __forceinline__ float dpp_add(float v) {
    int t = __builtin_amdgcn_update_dpp(0, __float_as_int(v), CTRL,
                                        /*row_mask=*/0xF, /*bank_mask=*/0xF,
                                        /*bound_ctrl=*/true);
    return v + __int_as_float(t);
}

// Sum within each 16-lane half of a wave32, pure VALU (no LDS, no waits).
__device__ __forceinline__ float hred16(float v) {
    v = dpp_add<0xB1>(v);    // quad_perm [1,0,3,2]  (xor 1)
    v = dpp_add<0x4E>(v);    // quad_perm [2,3,0,1]  (xor 2)
    v = dpp_add<0x141>(v);   // row_half_mirror      (xor 4 on quad-sums)
    v = dpp_add<0x140>(v);   // row_mirror           (xor 8 on octet-sums)
    return v;
}

__launch_bounds__(256, 2)
__global__ void divenc_fused_kernel(const float* __restrict__ x,
                                    const float* __restrict__ W1,
                                    const float* __restrict__ b1,
                                    const float* __restrict__ gamma,
                                    const float* __restrict__ beta,
                                    const float* __restrict__ W2,
                                    const float* __restrict__ b2,
                                    float* __restrict__ out) {
    // Double-buffered 16x1024-float input tiles (2x64KB) + 8KB output tile.
    __shared__ float xs[2][16 * 1024];
    __shared__ float os[16 * Q_N];

    const int tid = threadIdx.x;
    const int t0  = blockIdx.x * TILES;   // first 16-row tile owned by block

    // Async prefetch of one 64KB tile straight into LDS (no VGPR staging).
    // Per thread: 16 x B128, XOR-swizzled float4 column (conflict-free for
    // the strided A-fragment reads). Low 32 bits of a generic LDS address
    // are the LDS byte offset (ISA 10.2), which is what VDST wants.
    auto issue_tile = [&](int tg, int buf) {
        const float* xblk = x + (size_t)tg * 16 * (Q_N * V_N);
        #pragma unroll
        for (int i = 0; i < 16; ++i) {
            unsigned long long ga =
                (unsigned long long)(uintptr_t)(xblk + i * 1024 + tid * 4);
            unsigned int la =
                (unsigned int)(uintptr_t)(&xs[buf][i * 1024 + ((tid ^ i) * 4)]);
            asm volatile("global_load_async_to_lds_b128 %0, %1, off"
                         :: "v"(la), "v"(ga) : "memory");
        }
    };

    const int lane = tid & 31;
    const int wave = tid >> 5;
    const int m    = lane & 15;   // M row (A) / N column (B,C,D) within half
    const int hi   = lane >> 4;   // which 16-lane half

    issue_tile(t0, 0);            // prologue: tile 0 -> buf 0

    for (int t = 0; t < TILES; ++t) {
        const int cur = t & 1;
        if (t + 1 < TILES) {
            // Prefetch next tile into the other buffer; safe: trailing
            // barrier of iteration t-1 guarantees nobody still reads it.
            issue_tile(t0 + t + 1, cur ^ 1);
            // Async loads complete in order: <=16 outstanding means all of
            // tile t's 16 per-thread loads have landed in LDS.
            asm volatile("s_wait_asynccnt 0x10" ::: "memory");
        } else {
            asm volatile("s_wait_asynccnt 0x0" ::: "memory");
        }
        __syncthreads();          // tile t visible to all waves

        const float* xbuf = xs[cur];

        for (int qi = 0; qi < 16; ++qi) {
            const int q = wave * 16 + qi;

            // ---- A fragments: 16x4 fp32, two K-steps covering V=8 ----
            const float* rowp = xbuf + m * 1024;
            const int c0 = (((q * 2 + 0) ^ m) * 4) + hi * 2;
            const int c1 = (((q * 2 + 1) ^ m) * 4) + hi * 2;
            v2f a0 = *reinterpret_cast<const v2f*>(rowp + c0);
            v2f a1 = *reinterpret_cast<const v2f*>(rowp + c1);

            // ---- B fragments: 4x16 fp32 per N-tile (H=32 -> two tiles) ----
            const float* w1q = W1 + q * (V_N * H_N);
            v2f b00, b01, b10, b11;
            b00.x = w1q[(hi * 2 + 0) * H_N + m];
            b00.y = w1q[(hi * 2 + 1) * H_N + m];
            b01.x = w1q[(hi * 2 + 0) * H_N + 16 + m];
            b01.y = w1q[(hi * 2 + 1) * H_N + 16 + m];
            b10.x = w1q[(4 + hi * 2 + 0) * H_N + m];
            b10.y = w1q[(4 + hi * 2 + 1) * H_N + m];
            b11.x = w1q[(4 + hi * 2 + 0) * H_N + 16 + m];
            b11.y = w1q[(4 + hi * 2 + 1) * H_N + 16 + m];

            // ---- h[16x32] = x_tile[16x8] @ W1[q][8x32] via 4 WMMA ops ----
            v8f acc0 = {};
            v8f acc1 = {};
            acc0 = __builtin_amdgcn_wmma_f32_16x16x4_f32(false, a0, false, b00,
                                                         (short)0, acc0, false, false);
            acc0 = __builtin_amdgcn_wmma_f32_16x16x4_f32(false, a1, false, b10,
                                                         (short)0, acc0, false, false);
            acc1 = __builtin_amdgcn_wmma_f32_16x16x4_f32(false, a0, false, b01,
                                                         (short)0, acc1, false, false);
            acc1 = __builtin_amdgcn_wmma_f32_16x16x4_f32(false, a1, false, b11,
                                                         (short)0, acc1, false, false);

            // ---- folded LN-affine + linear2 constants ----
            // dot(hn, w2) = rstd*(dot(h, g*w2) - mu*S_gw) + C_bw
            const int qh = q * H_N;
            const float bb0 = b1[qh + m],    bb1 = b1[qh + 16 + m];
            const float gw0 = gamma[qh + m]      * W2[qh + m];
            const float gw1 = gamma[qh + 16 + m] * W2[qh + 16 + m];
            const float S_gw = hred16(gw0 + gw1);
            const float C_bw = hred16(beta[qh + m]      * W2[qh + m] +
                                      beta[qh + 16 + m] * W2[qh + 16 + m]) + b2[q];

            // D layout: acc[r] -> row (hi ? 8+r : r), column m (+16 for acc1).
            #pragma unroll
            for (int r = 0; r < 8; ++r) {
                float h0 = acc0[r] + bb0;
                float h1 = acc1[r] + bb1;
                // ELU
                h0 = h0 > 0.f ? h0 : (__expf(h0) - 1.f);
                h1 = h1 > 0.f ? h1 : (__expf(h1) - 1.f);
                // Three independent 16-lane reductions (ILP across DPP chains)
                const float s  = hred16(h0 + h1);             // sum(h)
                const float ss = hred16(h0 * h0 + h1 * h1);   // sum(h^2)
                const float sd = hred16(h0 * gw0 + h1 * gw1); // dot(h, g*w2)
                const float mu   = s * (1.0f / 32.0f);
                const float var  = ss * (1.0f / 32.0f) - mu * mu;
                const float rstd = rsqrtf(var + LN_EPS);
                const float d = rstd * (sd - mu * S_gw) + C_bw;
                if (m == 0) {
                    os[(hi ? 8 + r : r) * Q_N + q] = d;   // stage in LDS
                }
            }
        }

        // All os writes done; also fences buf[cur] reads before it is
        // overwritten by the prefetch issued at iteration t+1.
        __syncthreads();

        // Coalesced flush: this tile's [16 x 128] output is contiguous.
        float* outbase = out + (size_t)(t0 + t) * 16 * Q_N;
        #pragma unroll
        for (int i = 0; i < 2; ++i) {
            const int f4 = i * 256 + tid;   // 512 float4s = 2048 floats
            *reinterpret_cast<v4f*>(outbase + f4 * 4) =
                *reinterpret_cast<const v4f*>(os + f4 * 4);
        }
        // os reuse is safe: next writes happen only after the wait+barrier
        // at the top of iteration t+1, by which time all flush reads issued.
    }
}

extern "C" void kernel_launch(void* const* d_in, const int* in_sizes, int n_in,
                              void* d_out, int out_size, void* d_ws, size_t ws_size,
                              hipStream_t stream) {
    (void)in_sizes; (void)n_in; (void)out_size; (void)d_ws; (void)ws_size;
    const float* x     = (const float*)d_in[0];
    const float* W1    = (const float*)d_in[1];
    const float* b1    = (const float*)d_in[2];
    const float* gamma = (const float*)d_in[3];
    const float* beta  = (const float*)d_in[4];
    const float* W2    = (const float*)d_in[5];
    const float* b2    = (const float*)d_in[6];
    float* out = (float*)d_out;

    dim3 grid(B_TOT / (16 * TILES));  // 512 blocks, 4 double-buffered tiles each
    dim3 block(256);                  // 8 waves; each wave owns 16 q slices
    divenc_fused_kernel<<<grid, block, 0, stream>>>(x, W1, b1, gamma, beta, W2, b2, out);
}